// TreeSimpleGRU_61443802136746
// MI455X (gfx1250) — compile-verified
//
#include <hip/hip_runtime.h>
#include <hip/hip_bf16.h>

typedef __attribute__((ext_vector_type(2))) float v2f;
typedef __attribute__((ext_vector_type(8))) float v8f;

#define N_NODES    50000
#define IN_DIM     256
#define MEM_DIM    128
#define GATES      384      // 3 * MEM_DIM
#define NUM_LEVELS 8

__device__ __forceinline__ v2f ldv2(const float* p) { return *(const v2f*)p; }

__device__ __forceinline__ v8f wmma_f32(v2f a, v2f b, v8f c) {
  // D = A(16x4) * B(4x16) + C, full f32 — emits v_wmma_f32_16x16x4_f32
  return __builtin_amdgcn_wmma_f32_16x16x4_f32(false, a, false, b, (short)0, c,
                                               false, false);
}

__device__ __forceinline__ float sigmoid_f(float x) {
  return 1.0f / (1.0f + __expf(-x));
}

// ---------------------------------------------------------------------------
// Zero-fill the agg accumulator (must be re-zeroed every call: harness does
// not re-poison between replays and we must not carry state across calls).
// ---------------------------------------------------------------------------
__global__ void zero_f32_kernel(float* __restrict__ p, int n) {
  int i = blockIdx.x * blockDim.x + threadIdx.x;
  if (i < n) p[i] = 0.0f;
}

// ---------------------------------------------------------------------------
// gi[N,384] = w_emb[N,256] @ W_ih[384,256]^T + b_ih
// Block = 8 waves, 16-node stripe. Wave w computes gate tiles at columns
// {w*16, 128+w*16, 256+w*16}; A fragment shared across the 3 WMMA per k-step.
// ---------------------------------------------------------------------------
__global__ __launch_bounds__(256) void gi_gemm_kernel(
    const float* __restrict__ w_emb, const float* __restrict__ W_ih,
    const float* __restrict__ b_ih, float* __restrict__ gi) {
  const int wave = threadIdx.x >> 5;
  const int lane = threadIdx.x & 31;
  const int half = lane >> 4;          // 0: K={k,k+1}, 1: K={k+2,k+3}
  const int l15  = lane & 15;
  const int r0   = blockIdx.x * 16;    // node stripe base (50000/16 exact)
  const int j0   = wave * 16 + l15;    // this lane's output column (gate r)

  const float* Arow = w_emb + (size_t)(r0 + l15) * IN_DIM + 2 * half;
  const float* B0   = W_ih + (size_t)(j0)       * IN_DIM + 2 * half;
  const float* B1   = W_ih + (size_t)(j0 + 128) * IN_DIM + 2 * half;
  const float* B2   = W_ih + (size_t)(j0 + 256) * IN_DIM + 2 * half;

  v8f acc0 = {}, acc1 = {}, acc2 = {};
#pragma unroll 4
  for (int k = 0; k < IN_DIM; k += 4) {
    v2f a = ldv2(Arow + k);
    acc0 = wmma_f32(a, ldv2(B0 + k), acc0);
    acc1 = wmma_f32(a, ldv2(B1 + k), acc1);
    acc2 = wmma_f32(a, ldv2(B2 + k), acc2);
  }

  const float bi0 = b_ih[j0], bi1 = b_ih[j0 + 128], bi2 = b_ih[j0 + 256];
#pragma unroll
  for (int i = 0; i < 8; ++i) {
    const int row = r0 + i + 8 * half;          // D layout: M = i + 8*half
    float* g = gi + (size_t)row * GATES;
    g[j0]       = acc0[i] + bi0;
    g[j0 + 128] = acc1[i] + bi1;
    g[j0 + 256] = acc2[i] + bi2;
  }
}

// ---------------------------------------------------------------------------
// One tree level: for the 16-node stripe (skipped unless it contains a node
// at `lvl`), compute gh = agg @ W_hh^T + b_hh with f32 WMMA, then per masked
// node: r/z/n gates, h_new, store h, atomic-add h_new into agg[parent].
// ---------------------------------------------------------------------------
__global__ __launch_bounds__(256) void level_step_kernel(
    const float* __restrict__ W_hh, const float* __restrict__ b_hh,
    const float* __restrict__ gi, const int* __restrict__ parent,
    const int* __restrict__ level, float* __restrict__ agg,
    float* __restrict__ h_out, int lvl) {
  const int wave = threadIdx.x >> 5;
  const int lane = threadIdx.x & 31;
  const int half = lane >> 4;
  const int l15  = lane & 15;
  const int r0   = blockIdx.x * 16;

  // Levels are contiguous blocks of node indices -> most stripes exit here.
  const int lv_probe = level[r0 + l15];
  if (__ballot(lv_probe == lvl) == 0ull) return;   // uniform across waves

  const int c = wave * 16 + l15;                   // mem column for this lane
  const float* Arow = agg  + (size_t)(r0 + l15) * MEM_DIM + 2 * half;
  const float* Br   = W_hh + (size_t)(c)        * MEM_DIM + 2 * half;
  const float* Bz   = W_hh + (size_t)(c + 128)  * MEM_DIM + 2 * half;
  const float* Bn   = W_hh + (size_t)(c + 256)  * MEM_DIM + 2 * half;

  v8f ar = {}, az = {}, an = {};
#pragma unroll 4
  for (int k = 0; k < MEM_DIM; k += 4) {
    v2f a = ldv2(Arow + k);
    ar = wmma_f32(a, ldv2(Br + k), ar);
    az = wmma_f32(a, ldv2(Bz + k), az);
    an = wmma_f32(a, ldv2(Bn + k), an);
  }

  const float br = b_hh[c], bz = b_hh[c + 128], bn = b_hh[c + 256];
#pragma unroll
  for (int i = 0; i < 8; ++i) {
    const int node = r0 + i + 8 * half;
    if (level[node] != lvl) continue;              // after WMMAs: EXEC was full
    const float* g  = gi + (size_t)node * GATES;
    const float  hc = agg[(size_t)node * MEM_DIM + c];
    const float  r  = sigmoid_f(g[c]       + ar[i] + br);
    const float  z  = sigmoid_f(g[c + 128] + az[i] + bz);
    const float  nn = tanhf(g[c + 256] + r * (an[i] + bn));
    const float  hv = (1.0f - z) * nn + z * hc;
    h_out[(size_t)node * MEM_DIM + c] = hv;        // each node written once
    unsafeAtomicAdd(agg + (size_t)parent[node] * MEM_DIM + c, hv);
  }
}

// ---------------------------------------------------------------------------
extern "C" void kernel_launch(void* const* d_in, const int* in_sizes, int n_in,
                              void* d_out, int out_size, void* d_ws,
                              size_t ws_size, hipStream_t stream) {
  const float* w_emb  = (const float*)d_in[0];   // [50000,256]
  const float* W_ih   = (const float*)d_in[1];   // [384,256]
  const float* W_hh   = (const float*)d_in[2];   // [384,128]
  const float* b_ih   = (const float*)d_in[3];   // [384]
  const float* b_hh   = (const float*)d_in[4];   // [384]
  const int*   parent = (const int*)d_in[5];     // [50000]
  const int*   level  = (const int*)d_in[6];     // [50000]
  float* h_out = (float*)d_out;                  // [50000,128]

  float* gi  = (float*)d_ws;                              // 50000*384 f32
  float* agg = gi + (size_t)N_NODES * GATES;              // 50001*128 f32
  const int agg_n = (N_NODES + 1) * MEM_DIM;

  zero_f32_kernel<<<(agg_n + 255) / 256, 256, 0, stream>>>(agg, agg_n);

  const dim3 grid(N_NODES / 16);   // 3125 stripes, exact
  const dim3 block(256);           // 8 wave32s
  gi_gemm_kernel<<<grid, block, 0, stream>>>(w_emb, W_ih, b_ih, gi);

  for (int l = 0; l < NUM_LEVELS; ++l) {
    level_step_kernel<<<grid, block, 0, stream>>>(W_hh, b_hh, gi, parent,
                                                  level, agg, h_out, l);
  }
}